// GNN_61125974556855
// MI455X (gfx1250) — compile-verified
//
#include <hip/hip_runtime.h>
#include <cstdint>

#define NPTS 8192
#define BATCH 2
#define COUT 64
#define KNN 20
#define NCLS 40
#define BLK 256

typedef __attribute__((ext_vector_type(2))) float v2f;
typedef __attribute__((ext_vector_type(8))) float v8f;
typedef int v4i __attribute__((vector_size(16)));          // b128 payload type
typedef __attribute__((address_space(1))) v4i* gptr_v4i;   // global (device)
typedef __attribute__((address_space(3))) v4i* lptr_v4i;   // LDS (shared)

#if defined(__has_builtin)
#if __has_builtin(__builtin_amdgcn_global_load_async_to_lds_b128) && \
    __has_builtin(__builtin_amdgcn_s_wait_asynccnt)
#define HAVE_ASYNC_LDS 1
#endif
#endif
#ifndef HAVE_ASYNC_LDS
#define HAVE_ASYNC_LDS 0
#endif

// Exact 64-bit key: (bits(d^2) << 32) | m.  d^2 >= 0 so float order == bit
// order; low word reproduces lax.top_k's lower-index tie-break.
#define TRY_INSERT(xmval, midx)                                               \
    {                                                                         \
        float d_  = (xmval) - xn;                                             \
        float d2_ = d_ * d_;                                                  \
        if (d2_ <= t19) {        /* 32-bit fast guard; ties fall through */   \
            unsigned long long key_ =                                         \
                ((unsigned long long)__float_as_uint(d2_) << 32) |            \
                (unsigned)(midx);                                             \
            _Pragma("unroll")                                                 \
            for (int j_ = 0; j_ < KNN; ++j_) {                                \
                unsigned long long lo_ = keys[j_] < key_ ? keys[j_] : key_;   \
                unsigned long long hi_ = keys[j_] < key_ ? key_ : keys[j_];   \
                keys[j_] = lo_; key_ = hi_;                                   \
            }                                                                 \
            t19 = __uint_as_float((unsigned)(keys[KNN - 1] >> 32));           \
        }                                                                     \
    }

// ---------------------------------------------------------------------------
// Kernel A: per-point 1-D top-20 selection + collapsed EdgeConv+BN+ReLU+k-max,
// with block-level max/sum reduction over n into workspace accumulators.
// ---------------------------------------------------------------------------
__global__ __launch_bounds__(BLK) void knn_edge_kernel(
    const float* __restrict__ x,      const float* __restrict__ conv_w,
    const float* __restrict__ bn_g,   const float* __restrict__ bn_b,
    const float* __restrict__ bn_m,   const float* __restrict__ bn_v,
    float* __restrict__ ws_max,       float* __restrict__ ws_sum)
{
    __shared__ __align__(16) float xs[NPTS];         // 32 KB: whole batch row
    __shared__ float dminS[BLK], dmaxS[BLK], xnS[BLK];
    __shared__ float redM[BLK], redS[BLK];

    const int tid  = threadIdx.x;
    const int bpb  = NPTS / BLK;                     // blocks per batch = 32
    const int b    = blockIdx.x / bpb;
    const int n0   = (blockIdx.x % bpb) * BLK;
    const float* xb = x + b * NPTS;

    __builtin_prefetch(&xb[tid * 4], 0, 3);          // global_prefetch_b8

#if HAVE_ASYNC_LDS
    // ASYNCcnt-tracked DMA: global -> LDS, no VGPR round trip.
    {
        const float4* g4 = (const float4*)xb;
        for (int i = tid; i < NPTS / 4; i += BLK) {
            __builtin_amdgcn_global_load_async_to_lds_b128(
                (gptr_v4i)(g4 + i),
                (lptr_v4i)((float4*)xs + i),
                0, 0);
        }
        __builtin_amdgcn_s_wait_asynccnt(0);         // this wave's copies done
    }
#else
    {
        const float4* g4 = (const float4*)xb;
        float4*       s4 = (float4*)xs;
        for (int i = tid; i < NPTS / 4; i += BLK) s4[i] = g4[i];
    }
#endif
    __syncthreads();                                 // cross-wave visibility

    const int   n  = n0 + tid;
    const float xn = xs[n];

    unsigned long long keys[KNN];
#pragma unroll
    for (int j = 0; j < KNN; ++j)
        keys[j] = (0x7F800000ull << 32) | 0xFFFFFFFFull;   // +inf d^2 sentinel
    float t19 = __uint_as_float(0x7F800000u);              // +inf threshold

    const float4* xs4 = (const float4*)xs;
    for (int m4 = 0; m4 < NPTS; m4 += 4) {           // 1 ds_load_b128 / 4 cand
        float4 xm = xs4[m4 >> 2];                    // wave-uniform broadcast
        TRY_INSERT(xm.x, m4 + 0);
        TRY_INSERT(xm.y, m4 + 1);
        TRY_INSERT(xm.z, m4 + 2);
        TRY_INSERT(xm.w, m4 + 3);
    }

    // EdgeConv is linear in d_k: only dmin/dmax over the neighbor set matter.
    float dmin = 0.0f, dmax = 0.0f;                  // self (d=0) always in set
#pragma unroll
    for (int j = 0; j < KNN; ++j) {
        unsigned m = (unsigned)keys[j];
        float dk = xs[m] - xn;
        dmin = fminf(dmin, dk);
        dmax = fmaxf(dmax, dk);
    }
    dminS[tid] = dmin; dmaxS[tid] = dmax; xnS[tid] = xn;
    __syncthreads();

    // phase 2: h[o,n] = relu(a*(a>=0?dmax:dmin) + c0*xn + shift), reduce over n
    const int o     = tid & (COUT - 1);
    const int chunk = tid >> 6;                      // 0..3, 64 points each
    const float w0    = conv_w[o * 2 + 0];
    const float w1    = conv_w[o * 2 + 1];
    const float scale = bn_g[o] * rsqrtf(bn_v[o] + 1e-5f);
    const float shift = bn_b[o] - bn_m[o] * scale;
    const float a  = w0 * scale;
    const float c0 = w1 * scale;

    float lmax = 0.0f, lsum = 0.0f;                  // relu >= 0 -> 0 is safe id
    const int base = chunk * 64;
#pragma unroll 4
    for (int i = 0; i < 64; ++i) {
        int   p = base + i;
        float d = (a >= 0.0f) ? dmaxS[p] : dminS[p];
        float v = fmaf(a, d, fmaf(c0, xnS[p], shift));
        v = fmaxf(v, 0.0f);
        lmax = fmaxf(lmax, v);
        lsum += v;
    }
    redM[tid] = lmax; redS[tid] = lsum;
    __syncthreads();

    if (chunk == 0) {
        float m0 = fmaxf(fmaxf(redM[o], redM[o + 64]),
                         fmaxf(redM[o + 128], redM[o + 192]));
        float s0 = redS[o] + redS[o + 64] + redS[o + 128] + redS[o + 192];
        // relu output >= 0: integer max on float bits is order-preserving
        atomicMax((unsigned int*)&ws_max[b * COUT + o], __float_as_uint(m0));
        atomicAdd(&ws_sum[b * COUT + o], s0);
    }
}

// ---------------------------------------------------------------------------
// Kernel B: classifier head pooled(2x128) x lin_w^T(128x40) on
// V_WMMA_F32_16X16X4_F32 (exact f32). One wave, 3 N-tiles x 32 K-steps.
// ---------------------------------------------------------------------------
__device__ __forceinline__ float pooled_val(const float* __restrict__ wsm,
                                            const float* __restrict__ wss,
                                            int m, int j)
{
    int   mr   = (m < BATCH) ? m : 0;                // clamp (no OOB)
    float vmax = wsm[mr * COUT + (j & 63)];
    float vsum = wss[mr * COUT + (j & 63)] * (1.0f / NPTS);
    float v    = (j < COUT) ? vmax : vsum;
    return (m < BATCH) ? v : 0.0f;                   // zero pad rows 2..15
}

__global__ __launch_bounds__(32) void classifier_wmma(
    const float* __restrict__ lin_w,
    const float* __restrict__ ws_max, const float* __restrict__ ws_sum,
    float* __restrict__ out)
{
    const int lane  = threadIdx.x;                   // 0..31, EXEC all ones
    const int mrow  = lane & 15;                     // A: M = lane%16
    const int khalf = lane >> 4;                     // 0 -> K{0,1}, 1 -> K{2,3}

    for (int nt = 0; nt < 3; ++nt) {                 // class tiles 0..47
        v8f c = {};
        for (int k0 = 0; k0 < 2 * COUT; k0 += 4) {   // K = 128
            const int ka = k0 + khalf * 2;
            v2f av, bv;
            // A 16x4 layout: lanes 0-15 vgpr{0,1}=K{0,1}; lanes 16-31 =K{2,3}
            av.x = pooled_val(ws_max, ws_sum, mrow, ka + 0);
            av.y = pooled_val(ws_max, ws_sum, mrow, ka + 1);
            // B 4x16 layout: lane holds N=lane%16, same K-half split
            const int cls  = nt * 16 + mrow;         // N index = class
            const int clsc = (cls < NCLS) ? cls : 0; // clamp (no OOB)
            const float bm = (cls < NCLS) ? 1.0f : 0.0f;
            bv.x = lin_w[clsc * 2 * COUT + ka + 0] * bm;
            bv.y = lin_w[clsc * 2 * COUT + ka + 1] * bm;
            c = __builtin_amdgcn_wmma_f32_16x16x4_f32(
                    false, av, false, bv, (short)0, c, false, false);
        }
        // C 16x16 f32 layout: lanes 0-15 vgpr r -> M=r; lanes 16-31 -> M=8+r
        const int mBase = (lane < 16) ? 0 : 8;
#pragma unroll
        for (int r = 0; r < 8; ++r) {
            int row = mBase + r;
            int cls = nt * 16 + mrow;
            if (row < BATCH && cls < NCLS)
                out[row * NCLS + cls] = c[r];
        }
    }
}

// ---------------------------------------------------------------------------
extern "C" void kernel_launch(void* const* d_in, const int* in_sizes, int n_in,
                              void* d_out, int out_size, void* d_ws, size_t ws_size,
                              hipStream_t stream)
{
    const float* x      = (const float*)d_in[0];
    const float* conv_w = (const float*)d_in[1];
    const float* bn_g   = (const float*)d_in[2];
    const float* bn_b   = (const float*)d_in[3];
    const float* bn_m   = (const float*)d_in[4];
    const float* bn_v   = (const float*)d_in[5];
    const float* lin_w  = (const float*)d_in[6];
    float*       out    = (float*)d_out;

    float* wsf    = (float*)d_ws;
    float* ws_max = wsf;                       // [BATCH*COUT]
    float* ws_sum = wsf + BATCH * COUT;        // [BATCH*COUT]

    (void)hipMemsetAsync(d_ws, 0, 2 * BATCH * COUT * sizeof(float), stream);

    knn_edge_kernel<<<BATCH * (NPTS / BLK), BLK, 0, stream>>>(
        x, conv_w, bn_g, bn_b, bn_m, bn_v, ws_max, ws_sum);

    classifier_wmma<<<1, 32, 0, stream>>>(lin_w, ws_max, ws_sum, out);
}